// ComplexGATLayer_70282844832247
// MI455X (gfx1250) — compile-verified
//
#include <hip/hip_runtime.h>
#include <math.h>

// ComplexGATLayer for MI455X (gfx1250, wave32).
// Pipeline: [init] -> [WMMA f32 GEMM h=x@W^T] -> [attention dots] ->
//           [edge logits + seg-max] -> [exp + seg-sum] -> [scatter aggregate] ->
//           [bias/phase-mag/LayerNorm/GELU].

typedef float v2f __attribute__((ext_vector_type(2)));
typedef float v8f __attribute__((ext_vector_type(8)));

#define HEADS 4
#define CCH   32
#define DIM   128   // IN_DIM == HEADS*CCH == 128

// ---------------------------------------------------------------- utilities
__device__ __forceinline__ void atomicMaxF32(float* addr, float val) {
  // Native int atomics (no CAS loop). Works for mixed-sign values with -inf init.
  if (val >= 0.0f) atomicMax((int*)addr, __float_as_int(val));
  else             atomicMin((unsigned int*)addr, __float_as_uint(val));
}

__device__ __forceinline__ float wave_sum32(float v) {
  #pragma unroll
  for (int off = 16; off > 0; off >>= 1) v += __shfl_xor(v, off, 32);
  return v;
}

// ---------------------------------------------------------------- init
__global__ void gat_init_kernel(float* __restrict__ out, float* __restrict__ seg_max,
                                float* __restrict__ seg_sum, int total_out, int nh) {
  int idx = blockIdx.x * blockDim.x + threadIdx.x;
  if (idx < total_out) out[idx] = 0.0f;
  if (idx < nh) { seg_max[idx] = -INFINITY; seg_sum[idx] = 0.0f; }
}

// ---------------------------------------------------------------- GEMM (WMMA)
// h[n, j] = sum_k x[n,k] * W[j,k].  One wave per 16-row stripe, 8 col tiles,
// K=128 in 32 steps of V_WMMA_F32_16X16X4_F32.  Persistent blocks (grid-stride
// over row tiles) amortize the 64 KB LDS staging of W.
//
// LDS layout ("pair-K + bank swizzle"): element W[n,k] lives at
//   Wt2[(k>>1)*256 + (((n + ((k>>1)&1)*16) & 127) << 1) + (k&1)]
// A B-fragment (K = k0+khalf, k0+khalf+1 ; N = n) is then ONE aligned 8-byte
// ds_load_b64 into an even VGPR pair (no repack movs before v_wmma), and since
// p = 2*kb + (lane>>4), the half-wave reading pair p+1 is rotated by 16 n-slots
// (32 dwords) -> the two half-waves hit disjoint bank halves (conflict-free).
__global__ void __launch_bounds__(128)
gat_gemm_kernel(const float* __restrict__ x, const float* __restrict__ W,
                float* __restrict__ h, int n_tiles) {
  __shared__ float Wt2[64 * 256];   // exactly 64 KB

  const int tid = threadIdx.x;
  // Contiguous (conflict-free) LDS writes; strided global reads of the 64 KB
  // W matrix stay resident in L0/L2 after the first block touches them.
  for (int i = tid; i < 64 * 256; i += 128) {
    int p   = i >> 8;                              // K pair index
    int d   = i & 255;
    int sub = d & 1;
    int n   = ((d >> 1) - ((p & 1) << 4)) & 127;   // undo swizzle
    Wt2[i] = W[n * DIM + (p << 1) + sub];
  }
  __syncthreads();

  const int wave  = tid >> 5;
  const int lane  = tid & 31;
  const int mrow  = lane & 15;          // M (A) / N-within-tile (B)
  const int khi   = lane >> 4;          // 0: lanes 0-15, 1: lanes 16-31
  const int khalf = khi << 1;           // K offset owned by this half-wave
  const int swz   = khi << 4;           // n-slot rotation for odd pair rows

  const v8f zero = {0.f, 0.f, 0.f, 0.f, 0.f, 0.f, 0.f, 0.f};

  for (int tile = blockIdx.x * 4 + wave; tile < n_tiles; tile += gridDim.x * 4) {
    const int row_base = tile * 16;

    v8f acc[8];
    #pragma unroll
    for (int t = 0; t < 8; ++t) acc[t] = zero;

    const float* xrow = x + (size_t)(row_base + mrow) * DIM + khalf;

    for (int kb = 0; kb < 32; ++kb) {
      const int k0 = kb * 4;
      // A fragment: 16x4 f32.  VGPR v holds K = k0 + khalf + v.
      const float2 av = *(const float2*)(xrow + k0);
      v2f a; a.x = av.x; a.y = av.y;

      const float* prow = &Wt2[((k0 >> 1) + khi) * 256];
      #pragma unroll
      for (int t = 0; t < 8; ++t) {
        const int n_swz = (t * 16 + mrow + swz) & 127;
        v2f b = *(const v2f*)(prow + (n_swz << 1));   // one ds_load_b64
        acc[t] = __builtin_amdgcn_wmma_f32_16x16x4_f32(
            false, a, false, b, (short)0, acc[t], false, false);
      }
    }

    // C/D layout: VGPR r holds M = r + 8*(lane/16), N = lane%16.
    const int rhalf = khi << 3;
    #pragma unroll
    for (int t = 0; t < 8; ++t) {
      #pragma unroll
      for (int r = 0; r < 8; ++r) {
        h[(size_t)(row_base + r + rhalf) * DIM + t * 16 + mrow] = acc[t][r];
      }
    }
  }
}

// ---------------------------------------------------------------- attention dots
__global__ void gat_att_kernel(const float* __restrict__ h,
                               const float* __restrict__ att_src,
                               const float* __restrict__ att_dst,
                               float* __restrict__ a_src, float* __restrict__ a_dst,
                               int nh) {
  int idx = blockIdx.x * blockDim.x + threadIdx.x;
  if (idx >= nh) return;
  int n = idx >> 2, hh = idx & 3;
  const float4* hp = (const float4*)(h + (size_t)n * DIM + hh * CCH);
  const float4* as = (const float4*)(att_src + hh * CCH);
  const float4* ad = (const float4*)(att_dst + hh * CCH);
  float s1 = 0.f, s2 = 0.f;
  #pragma unroll
  for (int i = 0; i < 8; ++i) {
    float4 hv = hp[i], v1 = as[i], v2 = ad[i];
    s1 += hv.x * v1.x + hv.y * v1.y + hv.z * v1.z + hv.w * v1.w;
    s2 += hv.x * v2.x + hv.y * v2.y + hv.z * v2.z + hv.w * v2.w;
  }
  a_src[idx] = s1;
  a_dst[idx] = s2;
}

// ---------------------------------------------------------------- edge pass 1
__global__ void gat_edge1_kernel(const int* __restrict__ src, const int* __restrict__ dst,
                                 const float* __restrict__ a_src, const float* __restrict__ a_dst,
                                 float* __restrict__ e_buf, float* __restrict__ seg_max,
                                 int eh) {
  int idx = blockIdx.x * blockDim.x + threadIdx.x;
  if (idx >= eh) return;
  int e = idx >> 2, hh = idx & 3;
  int s = src[e], d = dst[e];
  float v = a_src[s * HEADS + hh] + a_dst[d * HEADS + hh];
  v = (v >= 0.0f) ? v : 0.2f * v;
  e_buf[idx] = v;
  atomicMaxF32(&seg_max[d * HEADS + hh], v);
}

// ---------------------------------------------------------------- edge pass 2
__global__ void gat_edge2_kernel(const int* __restrict__ dst,
                                 const float* __restrict__ seg_max,
                                 float* __restrict__ e_buf, float* __restrict__ seg_sum,
                                 int eh) {
  int idx = blockIdx.x * blockDim.x + threadIdx.x;
  if (idx >= eh) return;
  int e = idx >> 2, hh = idx & 3;
  int d = dst[e];
  float ex = expf(e_buf[idx] - seg_max[d * HEADS + hh]);
  e_buf[idx] = ex;
  atomicAdd(&seg_sum[d * HEADS + hh], ex);
}

// ---------------------------------------------------------------- aggregate
// One wave32 per edge; lane owns 4 channels (float4), head = lane>>3.
__global__ void gat_agg_kernel(const int* __restrict__ src, const int* __restrict__ dst,
                               const float* __restrict__ h, const float* __restrict__ e_buf,
                               const float* __restrict__ seg_sum, float* __restrict__ out,
                               int ne) {
  int edge = (blockIdx.x * blockDim.x + threadIdx.x) >> 5;
  int lane = threadIdx.x & 31;
  if (edge >= ne) return;
  int s = src[edge], d = dst[edge];
  int hh = lane >> 3;
  float alpha = e_buf[edge * HEADS + hh] / (seg_sum[d * HEADS + hh] + 1e-16f);
  float4 hv = ((const float4*)(h + (size_t)s * DIM))[lane];
  float* op = out + (size_t)d * DIM + lane * 4;
  atomicAdd(op + 0, hv.x * alpha);
  atomicAdd(op + 1, hv.y * alpha);
  atomicAdd(op + 2, hv.z * alpha);
  atomicAdd(op + 3, hv.w * alpha);
}

// ---------------------------------------------------------------- post
// out += bias; mag = sqrt((out*cos p)^2 + (out*sin p)^2 + 1e-12);
// LayerNorm over 128 channels; exact GELU.  One wave32 per node, in-place.
__global__ void gat_post_kernel(float* __restrict__ out,
                                const float* __restrict__ bias, const float* __restrict__ phase,
                                const float* __restrict__ gamma, const float* __restrict__ beta,
                                int n) {
  int node = (blockIdx.x * blockDim.x + threadIdx.x) >> 5;
  int lane = threadIdx.x & 31;
  if (node >= n) return;

  float4 o = ((float4*)(out + (size_t)node * DIM))[lane];
  float4 b = ((const float4*)bias)[lane];
  float4 p = ((const float4*)phase)[lane];
  o.x += b.x; o.y += b.y; o.z += b.z; o.w += b.w;

  float4 mag;
  {
    float c, s;
    c = cosf(p.x); s = sinf(p.x); mag.x = sqrtf(o.x*c*o.x*c + o.x*s*o.x*s + 1e-12f);
    c = cosf(p.y); s = sinf(p.y); mag.y = sqrtf(o.y*c*o.y*c + o.y*s*o.y*s + 1e-12f);
    c = cosf(p.z); s = sinf(p.z); mag.z = sqrtf(o.z*c*o.z*c + o.z*s*o.z*s + 1e-12f);
    c = cosf(p.w); s = sinf(p.w); mag.w = sqrtf(o.w*c*o.w*c + o.w*s*o.w*s + 1e-12f);
  }

  float mu = wave_sum32(mag.x + mag.y + mag.z + mag.w) * (1.0f / DIM);
  float dx = mag.x - mu, dy = mag.y - mu, dz = mag.z - mu, dw = mag.w - mu;
  float var = wave_sum32(dx*dx + dy*dy + dz*dz + dw*dw) * (1.0f / DIM);
  float rstd = rsqrtf(var + 1e-5f);

  float4 g  = ((const float4*)gamma)[lane];
  float4 be = ((const float4*)beta)[lane];
  float4 hn;
  hn.x = dx * rstd * g.x + be.x;
  hn.y = dy * rstd * g.y + be.y;
  hn.z = dz * rstd * g.z + be.z;
  hn.w = dw * rstd * g.w + be.w;

  const float inv_sqrt2 = 0.70710678118654752f;
  float4 r;
  r.x = 0.5f * hn.x * (1.0f + erff(hn.x * inv_sqrt2));
  r.y = 0.5f * hn.y * (1.0f + erff(hn.y * inv_sqrt2));
  r.z = 0.5f * hn.z * (1.0f + erff(hn.z * inv_sqrt2));
  r.w = 0.5f * hn.w * (1.0f + erff(hn.w * inv_sqrt2));

  ((float4*)(out + (size_t)node * DIM))[lane] = r;
}

// ---------------------------------------------------------------- launch
extern "C" void kernel_launch(void* const* d_in, const int* in_sizes, int n_in,
                              void* d_out, int out_size, void* d_ws, size_t ws_size,
                              hipStream_t stream) {
  const float* x       = (const float*)d_in[0];
  const int*   ei      = (const int*)d_in[1];   // int32 (JAX default config)
  const float* W       = (const float*)d_in[2];
  const float* att_src = (const float*)d_in[3];
  const float* att_dst = (const float*)d_in[4];
  const float* bias    = (const float*)d_in[5];
  const float* phase   = (const float*)d_in[6];
  const float* gammap  = (const float*)d_in[7];
  const float* betap   = (const float*)d_in[8];
  float* out = (float*)d_out;

  const int n = in_sizes[0] / DIM;      // 100000
  const int e = in_sizes[1] / 2;        // 1600000
  const int* src = ei;
  const int* dst = ei + e;

  // Workspace carve-out (~80 MB): h | a_src | a_dst | seg_max | seg_sum | e_buf
  char* ws = (char*)d_ws;
  size_t off = 0;
  float* h       = (float*)(ws + off); off += (size_t)n * DIM * sizeof(float);
  float* a_src   = (float*)(ws + off); off += (size_t)n * HEADS * sizeof(float);
  float* a_dst   = (float*)(ws + off); off += (size_t)n * HEADS * sizeof(float);
  float* seg_max = (float*)(ws + off); off += (size_t)n * HEADS * sizeof(float);
  float* seg_sum = (float*)(ws + off); off += (size_t)n * HEADS * sizeof(float);
  float* e_buf   = (float*)(ws + off); off += (size_t)e * HEADS * sizeof(float);

  const int total_out = n * DIM;
  const int nh = n * HEADS;
  const int eh = e * HEADS;

  gat_init_kernel<<<(total_out + 255) / 256, 256, 0, stream>>>(
      out, seg_max, seg_sum, total_out, nh);

  const int n_tiles = n / 16;           // N == 100000 is divisible by 16
  gat_gemm_kernel<<<512, 128, 0, stream>>>(x, W, h, n_tiles);

  gat_att_kernel<<<(nh + 255) / 256, 256, 0, stream>>>(
      h, att_src, att_dst, a_src, a_dst, nh);

  gat_edge1_kernel<<<(eh + 255) / 256, 256, 0, stream>>>(
      src, dst, a_src, a_dst, e_buf, seg_max, eh);

  gat_edge2_kernel<<<(eh + 255) / 256, 256, 0, stream>>>(
      dst, seg_max, e_buf, seg_sum, eh);

  gat_agg_kernel<<<(e + 7) / 8, 256, 0, stream>>>(
      src, dst, h, e_buf, seg_sum, out, e);

  gat_post_kernel<<<(n + 7) / 8, 256, 0, stream>>>(
      out, bias, phase, gammap, betap, n);
}